// DetFocalLoss_16810501997096
// MI455X (gfx1250) — compile-verified
//
#include <hip/hip_runtime.h>
#include <stdint.h>

typedef float v2f __attribute__((ext_vector_type(2)));
typedef float v4f __attribute__((ext_vector_type(4)));
typedef float v8f __attribute__((ext_vector_type(8)));

#define ALPHA 0.25f
#define CLS_EPS 1e-4f
#define MAXM 64

// ---------------------------------------------------------------------------
// Exact 32-lane wave sum via V_WMMA_F32_16X16X4_F32.
// A (16x4): VGPR0 -> K=0 (lanes 0-15) / K=2 (lanes 16-31); VGPR1 -> K=1/K=3.
// Setting a.y = 0 zeroes K=1,3. With B = all-ones, D[m][n] = p_m + p_{m+16}.
// D VGPR v, lanes 0-15 hold row v, lanes 16-31 hold row v+8, so summing the
// 8 D registers per lane + one xor-16 shuffle yields the 32-lane total.
// Requires EXEC == all 1s: call uniformly from full waves only.
// ---------------------------------------------------------------------------
__device__ __forceinline__ float wmma_wave_reduce(float p) {
    v2f a; a.x = p;   a.y = 0.0f;
    v2f b; b.x = 1.0f; b.y = 1.0f;
    v8f c = {};
    c = __builtin_amdgcn_wmma_f32_16x16x4_f32(false, a, false, b,
                                              (short)0, c, false, false);
    float s = c[0] + c[1] + c[2] + c[3] + c[4] + c[5] + c[6] + c[7];
    s += __shfl_xor(s, 16, 32);
    return s;
}

__global__ void zero_kernel(float* p, int n) {
    int i = blockIdx.x * blockDim.x + threadIdx.x;
    if (i < n) p[i] = 0.0f;
}

// ---------------------------------------------------------------------------
// Kernel 1: per-anchor IoU vs M boxes -> argmax, pos/neg flag, label (packed
// u32), plus per-image num_pos, smooth-L1 reg sum, and embedding segment
// counts/sums. All accumulators aggregated in LDS, flushed once per block.
// ---------------------------------------------------------------------------
__global__ void iou_assign_kernel(const float* __restrict__ anchors,
                                  const float* __restrict__ regressions,
                                  const float* __restrict__ annotations,
                                  uint32_t* __restrict__ packed,
                                  float* __restrict__ stats,   // [B][4]: np, reg, cls
                                  float* __restrict__ counts,  // [B][M]
                                  float* __restrict__ sums,    // [B][M][3]
                                  int A, int M) {
    const int b = blockIdx.y;
    const int a = blockIdx.x * blockDim.x + threadIdx.x;
    __shared__ float s_ann[MAXM * 5];
    __shared__ float s_cnt[MAXM];
    __shared__ float s_sum[MAXM * 3];
    __shared__ float s_np, s_reg;

    if (threadIdx.x == 0) { s_np = 0.0f; s_reg = 0.0f; }
    for (int j = threadIdx.x; j < M; j += blockDim.x) {
        s_cnt[j] = 0.0f;
        s_sum[3 * j] = 0.0f; s_sum[3 * j + 1] = 0.0f; s_sum[3 * j + 2] = 0.0f;
    }
    for (int j = threadIdx.x; j < M * 5; j += blockDim.x)
        s_ann[j] = annotations[(size_t)b * M * 5 + j];
    __syncthreads();

    if (a < A) {
        // anchors are [y1, x1, y2, x2]
        const v4f an  = *(const v4f*)(anchors + 4 * (size_t)a);
        const float ay1 = an.x, ax1 = an.y, ay2 = an.z, ax2 = an.w;
        const float aw = ax2 - ax1, ah = ay2 - ay1;
        const float acx = ax1 + 0.5f * aw, acy = ay1 + 0.5f * ah;
        const float area_a = ah * aw;

        float best = -1.0f; int arg = 0;
        for (int j = 0; j < M; j++) {
            // boxes are [x1, y1, x2, y2, label]
            const float bx1 = s_ann[5 * j + 0], by1 = s_ann[5 * j + 1];
            const float bx2 = s_ann[5 * j + 2], by2 = s_ann[5 * j + 3];
            float iw = fmaxf(fminf(ax2, bx2) - fmaxf(ax1, bx1), 0.0f);
            float ih = fmaxf(fminf(ay2, by2) - fmaxf(ay1, by1), 0.0f);
            const float inter  = iw * ih;
            const float area_b = (bx2 - bx1) * (by2 - by1);
            const float ua  = fmaxf(area_a + area_b - inter, 1e-8f);
            const float iou = inter / ua;
            if (iou > best) { best = iou; arg = j; }   // first-max semantics
        }
        const int pos  = best >= 0.5f;
        const int flag = pos ? 1 : (best < 0.4f ? 2 : 0);
        const int label = (int)s_ann[5 * arg + 4];
        packed[(size_t)b * A + a] =
            (uint32_t)(arg & 0xff) | ((uint32_t)flag << 8) |
            ((uint32_t)(label & 0xff) << 10);

        if (pos) {
            const float bx1 = s_ann[5 * arg + 0], by1 = s_ann[5 * arg + 1];
            const float bx2 = s_ann[5 * arg + 2], by2 = s_ann[5 * arg + 3];
            float gw = bx2 - bx1, gh = by2 - by1;
            const float gcx = bx1 + 0.5f * gw, gcy = by1 + 0.5f * gh;
            gw = fmaxf(gw, 1.0f); gh = fmaxf(gh, 1.0f);
            const float rt[4] = { (gcy - acy) / ah, (gcx - acx) / aw,
                                  logf(gh / ah),    logf(gw / aw) };
            const float* r = regressions + ((size_t)b * A + a) * 7;
            float rs = 0.0f;
            #pragma unroll
            for (int k = 0; k < 4; k++) {
                const float d = fabsf(rt[k] - r[k]);
                rs += (d <= (1.0f / 9.0f)) ? 4.5f * d * d : d - (0.5f / 9.0f);
            }
            atomicAdd(&s_np, 1.0f);
            atomicAdd(&s_reg, rs);
            atomicAdd(&s_cnt[arg], 1.0f);
            atomicAdd(&s_sum[3 * arg + 0], r[4]);
            atomicAdd(&s_sum[3 * arg + 1], r[5]);
            atomicAdd(&s_sum[3 * arg + 2], r[6]);
        }
    }
    __syncthreads();
    if (threadIdx.x == 0) {
        atomicAdd(&stats[b * 4 + 0], s_np);
        atomicAdd(&stats[b * 4 + 1], s_reg);
    }
    for (int j = threadIdx.x; j < M; j += blockDim.x) {
        if (s_cnt[j] != 0.0f) {
            atomicAdd(&counts[b * M + j], s_cnt[j]);
            atomicAdd(&sums[(b * M + j) * 3 + 0], s_sum[3 * j + 0]);
            atomicAdd(&sums[(b * M + j) * 3 + 1], s_sum[3 * j + 1]);
            atomicAdd(&sums[(b * M + j) * 3 + 2], s_sum[3 * j + 2]);
        }
    }
}

// ---------------------------------------------------------------------------
// Kernel 2: streaming focal loss over B*A*C scores (128 MB; the BW-bound
// pass). Fully branchless core: for every kept element the term is
//   af * (1-y)^2 * (-log y),  y = is_target ? x : 1-x,
//   af = is_target ? 0.25 : (ignored ? 0 : 0.75)
// One v_log_f32 + one v_cndmask per element, no exec-mask churn.
// CT > 0 specializes C at compile time (div -> magic multiply) and streams
// 8 floats (two NT b128 loads) per iteration; CT == 0 is the generic
// 4-wide fallback (requires C % 4 == 0).
// ---------------------------------------------------------------------------
template <int CT>
__global__ void cls_focal_kernel(const float* __restrict__ cls,
                                 const uint32_t* __restrict__ packed,
                                 float* __restrict__ stats,
                                 int A, int Crt) {
    const int C = (CT > 0) ? CT : Crt;
    const int b = blockIdx.y;
    const v4f* base = (const v4f*)(cls + (size_t)b * A * C);
    const uint32_t* pk = packed + (size_t)b * A;

    float part = 0.0f;
    if (CT > 0) {
        const int n8 = (A * C) >> 3;               // C % 8 == 0 for this path
        for (int i = blockIdx.x * blockDim.x + threadIdx.x; i < n8;
             i += gridDim.x * blockDim.x) {
            const v4f v0 = __builtin_nontemporal_load(base + 2 * i);
            const v4f v1 = __builtin_nontemporal_load(base + 2 * i + 1);
            const int e = i << 3;
            const int a = e / C;                   // constant divisor
            const int c0 = e - a * C;
            const uint32_t p = pk[a];
            const uint32_t flag = (p >> 8) & 3;    // 0 ignore, 1 pos, 2 neg
            const int plab = (flag == 1) ? (int)((p >> 10) & 0xff) : -1;
            const float a0 = (flag != 0) ? (1.0f - ALPHA) : 0.0f;
            #pragma unroll
            for (int k = 0; k < 8; k++) {
                const float raw = (k < 4) ? v0[k] : v1[k - 4];
                const float x = fminf(fmaxf(raw, CLS_EPS), 1.0f - CLS_EPS);
                const bool is1 = (c0 + k) == plab;
                const float y  = is1 ? x : 1.0f - x;
                const float af = is1 ? ALPHA : a0;
                const float om = 1.0f - y;
                part = fmaf(af * om * om, -__logf(y), part);
            }
        }
    } else {
        const int n4 = (A * C) >> 2;
        for (int i = blockIdx.x * blockDim.x + threadIdx.x; i < n4;
             i += gridDim.x * blockDim.x) {
            const v4f v = __builtin_nontemporal_load(base + i);
            const int e = i << 2;
            const int a = e / C;
            const int c0 = e - a * C;
            const uint32_t p = pk[a];
            const uint32_t flag = (p >> 8) & 3;
            const int plab = (flag == 1) ? (int)((p >> 10) & 0xff) : -1;
            const float a0 = (flag != 0) ? (1.0f - ALPHA) : 0.0f;
            #pragma unroll
            for (int k = 0; k < 4; k++) {
                const float x = fminf(fmaxf(v[k], CLS_EPS), 1.0f - CLS_EPS);
                const bool is1 = (c0 + k) == plab;
                const float y  = is1 ? x : 1.0f - x;
                const float af = is1 ? ALPHA : a0;
                const float om = 1.0f - y;
                part = fmaf(af * om * om, -__logf(y), part);
            }
        }
    }

    const float w = wmma_wave_reduce(part);   // uniform call, EXEC all-1s
    __shared__ float s_w[32];
    const int wid = threadIdx.x >> 5, ln = threadIdx.x & 31;
    if (ln == 0) s_w[wid] = w;
    __syncthreads();
    if (threadIdx.x == 0) {
        float t = 0.0f;
        const int nw = blockDim.x >> 5;
        for (int i = 0; i < nw; i++) t += s_w[i];
        atomicAdd(&stats[b * 4 + 2], t);
    }
}

// ---------------------------------------------------------------------------
// Kernel 3: per-annotation embedding means -> workspace + output embs.
// ---------------------------------------------------------------------------
__global__ void emb_mean_kernel(const float* __restrict__ counts,
                                const float* __restrict__ sums,
                                float* __restrict__ mean,
                                float* __restrict__ out_embs, int BM) {
    const int i = blockIdx.x * blockDim.x + threadIdx.x;
    if (i < BM) {
        const float c = counts[i];
        const float inv = (c > 0.0f) ? 1.0f / fmaxf(c, 1.0f) : 0.0f;
        #pragma unroll
        for (int k = 0; k < 3; k++) {
            const float m = sums[3 * i + k] * inv;
            mean[3 * i + k] = m;
            out_embs[3 * i + k] = m;
        }
    }
}

// ---------------------------------------------------------------------------
// Kernel 4: second segment pass: sum ||emb - mean[argmax]||^2 over pos
// anchors per annotation.
// ---------------------------------------------------------------------------
__global__ void emb_sq_kernel(const float* __restrict__ regressions,
                              const uint32_t* __restrict__ packed,
                              const float* __restrict__ mean,
                              float* __restrict__ sq, int A, int M) {
    const int b = blockIdx.y;
    const int a = blockIdx.x * blockDim.x + threadIdx.x;
    if (a >= A) return;
    const uint32_t p = packed[(size_t)b * A + a];
    if (((p >> 8) & 3) == 1) {
        const int j = p & 0xff;
        const float* r = regressions + ((size_t)b * A + a) * 7 + 4;
        const float* m = mean + (b * M + j) * 3;
        const float d0 = r[0] - m[0], d1 = r[1] - m[1], d2 = r[2] - m[2];
        atomicAdd(&sq[b * M + j], d0 * d0 + d1 * d1 + d2 * d2);
    }
}

// ---------------------------------------------------------------------------
// Kernel 5: combine into the 3 scalar losses (trivial amount of work).
// ---------------------------------------------------------------------------
__global__ void finalize_kernel(const float* __restrict__ stats,
                                const float* __restrict__ counts,
                                const float* __restrict__ sq,
                                float* __restrict__ out, int B, int M) {
    if (threadIdx.x == 0 && blockIdx.x == 0) {
        float cl = 0.0f, rl = 0.0f, el = 0.0f;
        for (int b = 0; b < B; b++) {
            const float np = stats[b * 4 + 0];
            cl += stats[b * 4 + 2] / fmaxf(np, 1.0f);
            rl += stats[b * 4 + 1] / fmaxf(np * 4.0f, 1.0f);
            float s = 0.0f;
            for (int j = 0; j < M; j++) {
                const float c = counts[b * M + j];
                if (c > 0.0f) s += sq[b * M + j] / fmaxf(c * 3.0f, 1.0f);
            }
            el += s / (float)M;
        }
        out[0] = cl / (float)B;
        out[1] = (rl / (float)B) * 50.0f;
        out[2] = el / (float)B;
    }
}

extern "C" void kernel_launch(void* const* d_in, const int* in_sizes, int n_in,
                              void* d_out, int out_size, void* d_ws, size_t ws_size,
                              hipStream_t stream) {
    const float* classifications = (const float*)d_in[0];
    const float* regressions     = (const float*)d_in[1];
    const float* anchors         = (const float*)d_in[2];
    const float* annotations     = (const float*)d_in[3];

    const int A = in_sizes[2] / 4;           // anchors: (1, A, 4)
    const int B = in_sizes[1] / (7 * A);     // regressions: (B, A, 7)
    const int C = in_sizes[0] / (B * A);     // classifications: (B, A, C)
    const int M = in_sizes[3] / (B * 5);     // annotations: (B, M, 5)

    // Workspace layout
    uint32_t* packed = (uint32_t*)d_ws;                       // B*A
    float* stats  = (float*)(packed + (size_t)B * A);         // B*4
    float* counts = stats + B * 4;                            // B*M
    float* sums   = counts + B * M;                           // B*M*3
    float* sq     = sums + B * M * 3;                         // B*M
    float* mean   = sq + B * M;                               // B*M*3
    const int nz = B * 4 + B * M * 8;
    float* out = (float*)d_out;                               // [3 losses][B*M*3 embs]

    zero_kernel<<<(nz + 255) / 256, 256, 0, stream>>>(stats, nz);

    dim3 gA((A + 255) / 256, B);
    iou_assign_kernel<<<gA, 256, 0, stream>>>(anchors, regressions, annotations,
                                              packed, stats, counts, sums, A, M);

    dim3 gC(1024, B);
    if (C == 80) {
        cls_focal_kernel<80><<<gC, 256, 0, stream>>>(classifications, packed,
                                                     stats, A, C);
    } else {
        cls_focal_kernel<0><<<gC, 256, 0, stream>>>(classifications, packed,
                                                    stats, A, C);
    }

    emb_mean_kernel<<<(B * M + 127) / 128, 128, 0, stream>>>(counts, sums, mean,
                                                             out + 3, B * M);
    emb_sq_kernel<<<gA, 256, 0, stream>>>(regressions, packed, mean, sq, A, M);
    finalize_kernel<<<1, 32, 0, stream>>>(stats, counts, sq, out, B, M);
}